// HausdorffDTLoss_25108378812985
// MI455X (gfx1250) — compile-verified
//
#include <hip/hip_runtime.h>
#include <hip/hip_bf16.h>

// ---------------------------------------------------------------------------
// HausdorffDTLoss (ALPHA=2) for B=4, C=2, H=W=256.
//  loss = mean( (pred-gt)^2 * (pred_dt^2 + gt_dt^2) )
//  dt fields via exact brute-force separable squared-EDT (matches jnp ref
//  arithmetic: f in {0,1e9}, candidates f[j]+(i-j)^2 in f32, min, sqrt, ^2).
//  Reductions use V_WMMA_F32_16X16X4_F32 (A=data, B=ones) -> deterministic
//  f32 accumulation + gfx1250 matrix path.
// ---------------------------------------------------------------------------

typedef float v2f __attribute__((ext_vector_type(2)));
typedef float v8f __attribute__((ext_vector_type(8)));

#define HW 256
#define NB 4
#define NPLANES 16          // 4 fields (pred_fg, pred_bg, gt_fg, gt_bg) x 4 batches
#define NPIX (NB * HW * HW) // 262144 (mean denominator, B*1*H*W)

// Accumulate the sum of 64 f32 values (one 16x4 A-tile) into acc via WMMA.
// A layout (ISA 7.12.2, 32-bit A 16x4): lane l<16 -> A[l][0],A[l][1];
// lane l>=16 -> A[l-16][2],A[l-16][3].  Value index v = M*4+K.
// B = all-ones 4x16  =>  D[m][n] = sum_k A[m][k] + C[m][n].
__device__ __forceinline__ v8f wmma_tile_sum(const float* __restrict__ vals,
                                             int lane, v8f acc) {
  v2f a;
  if (lane < 16) {
    a.x = vals[lane * 4 + 0];
    a.y = vals[lane * 4 + 1];
  } else {
    a.x = vals[(lane - 16) * 4 + 2];
    a.y = vals[(lane - 16) * 4 + 3];
  }
  v2f b;
  b.x = 1.0f;
  b.y = 1.0f;
  return __builtin_amdgcn_wmma_f32_16x16x4_f32(
      /*neg_a=*/false, a, /*neg_b=*/false, b,
      /*c_mod=*/(short)0, acc, /*reuse_a=*/false, /*reuse_b=*/false);
}

// Extract total sum of all accumulated A-elements from the D accumulator.
// Lane n (n<16): VGPR r = D[r][n]    -> per-lane sum = rows 0..7
// Lane n+16   : VGPR r = D[r+8][n]   -> per-lane sum = rows 8..15
__device__ __forceinline__ float wmma_extract_total(v8f acc, int lane) {
  float s = acc[0] + acc[1] + acc[2] + acc[3] + acc[4] + acc[5] + acc[6] + acc[7];
  float o = __shfl(s, lane ^ 16, 32);
  return s + o; // identical on every lane
}

__global__ void hdt_init_flags(unsigned int* __restrict__ flags) {
  if (threadIdx.x < 8) flags[threadIdx.x] = 0u;
}

// Pass 1: 1D squared DT along W for each row of each of the 16 planes.
// One 256-thread block per (plane,row).  Exact emulation of
//   g[x] = min_j ( (mask[j] ? 1e9f : 0.f) + (x-j)^2 )   in f32.
__global__ void hdt_pass1(const float* __restrict__ mo,
                          const float* __restrict__ gt,
                          float* __restrict__ g,
                          unsigned int* __restrict__ flags) {
  __shared__ float f[HW];
  const int x = threadIdx.x;
  const int y = blockIdx.x & (HW - 1);
  const int p = blockIdx.x >> 8;      // plane
  const int field = p >> 2;           // 0:pred_fg 1:pred_bg 2:gt_fg 3:gt_bg
  const int b = p & 3;

  bool fg;
  if (field < 2) {
    // pred = argmax over C=2 -> 1 iff mo[c=1] > mo[c=0] (ties -> 0, like argmax)
    const float m0 = mo[((b * 2 + 0) * HW + y) * HW + x];
    const float m1 = mo[((b * 2 + 1) * HW + y) * HW + x];
    const bool pred = m1 > m0;
    fg = (field == 0) ? pred : !pred;
  } else {
    const float gv = gt[(b * HW + y) * HW + x];
    const bool gfg = gv > 0.5f;
    fg = (field == 2) ? gfg : !gfg;
  }
  f[x] = fg ? 1.0e9f : 0.0f;

  // has_fg guard flags (per batch, pred in [0..3], gt in [4..7])
  if (field == 0 || field == 2) {
    if (__any((int)fg) && (x & 31) == 0)
      atomicOr(&flags[(field == 0 ? 0 : 4) + b], 1u);
  }
  __syncthreads();

  float d = 3.0e38f;
#pragma unroll 4
  for (int j = 0; j < HW; ++j) {
    const int dx = x - j;
    d = fminf(d, f[j] + (float)(dx * dx));
  }
  g[(p * HW + y) * HW + x] = d;
}

// Pass 2: 1D squared DT along H for all 4 fields of one (batch, column),
// fused with err*dist and a WMMA block reduction -> one partial per block.
__global__ void hdt_pass2(const float* __restrict__ mo,
                          const float* __restrict__ gt,
                          const float* __restrict__ g,
                          const unsigned int* __restrict__ flags,
                          float* __restrict__ partials) {
  __shared__ float col[4][HW];
  __shared__ float red[HW];
  const int t = threadIdx.x;
  const int x = blockIdx.x & (HW - 1);
  const int b = blockIdx.x >> 8;

#pragma unroll
  for (int fld = 0; fld < 4; ++fld)
    col[fld][t] = g[(((fld * 4 + b) * HW + t) * HW) + x];
  __syncthreads();

  const int y = t;
  float dmin[4];
#pragma unroll
  for (int fld = 0; fld < 4; ++fld) {
    float d = 3.0e38f;
    for (int j = 0; j < HW; ++j) {
      const int dy = y - j;
      d = fminf(d, col[fld][j] + (float)(dy * dy));
    }
    dmin[fld] = d;
  }

  // field = sqrt(edt2_fg) + sqrt(edt2_bg) (one term is exactly 0); ^ALPHA(=2)
  float pred_dt = sqrtf(dmin[0]) + sqrtf(dmin[1]);
  float gt_dt = sqrtf(dmin[2]) + sqrtf(dmin[3]);
  if (flags[b] == 0u) pred_dt = 0.0f;      // no fg in pred plane
  if (flags[4 + b] == 0u) gt_dt = 0.0f;    // no fg in gt plane
  const float dist = pred_dt * pred_dt + gt_dt * gt_dt;

  const float m0 = mo[((b * 2 + 0) * HW + y) * HW + x];
  const float m1 = mo[((b * 2 + 1) * HW + y) * HW + x];
  const float pred = (m1 > m0) ? 1.0f : 0.0f;
  const float gv = gt[(b * HW + y) * HW + x];
  float e = pred - gv;
  e = e * e;

  red[t] = e * dist;
  __syncthreads();

  // wave 0 reduces 256 values = 4 WMMA 16x4 tiles (EXEC all-ones in wave 0)
  if (t < 32) {
    v8f acc = {};
#pragma unroll
    for (int tile = 0; tile < 4; ++tile)
      acc = wmma_tile_sum(&red[tile * 64], t, acc);
    const float tot = wmma_extract_total(acc, t);
    if (t == 0) partials[blockIdx.x] = tot;
  }
}

// Final deterministic reduction of 1024 partials (16 WMMA tiles) + mean.
__global__ void hdt_reduce(const float* __restrict__ partials,
                           float* __restrict__ out) {
  const int lane = threadIdx.x; // single wave32 block
  v8f acc = {};
#pragma unroll
  for (int tile = 0; tile < 16; ++tile)
    acc = wmma_tile_sum(partials + tile * 64, lane, acc);
  const float tot = wmma_extract_total(acc, lane);
  if (lane == 0) out[0] = tot / (float)NPIX;
}

extern "C" void kernel_launch(void* const* d_in, const int* in_sizes, int n_in,
                              void* d_out, int out_size, void* d_ws, size_t ws_size,
                              hipStream_t stream) {
  (void)in_sizes; (void)n_in; (void)out_size; (void)ws_size;
  const float* mo = (const float*)d_in[0]; // (4,2,256,256) f32
  const float* gt = (const float*)d_in[1]; // (4,1,256,256) f32

  float* g = (float*)d_ws;                              // 16*65536 f32 = 4 MB
  float* partials = g + NPLANES * HW * HW;              // 1024 f32
  unsigned int* flags = (unsigned int*)(partials + 1024); // 8 u32

  hdt_init_flags<<<1, 32, 0, stream>>>(flags);
  hdt_pass1<<<NPLANES * HW, HW, 0, stream>>>(mo, gt, g, flags);
  hdt_pass2<<<NB * HW, HW, 0, stream>>>(mo, gt, g, flags, partials);
  hdt_reduce<<<1, 32, 0, stream>>>(partials, (float*)d_out);
}